// KPConv_68126771249161
// MI455X (gfx1250) — compile-verified
//
#include <hip/hip_runtime.h>

// KPConv fused kernel for gfx1250 (MI455X), fp32 WMMA path.
//
// reference shapes: N=20000, Ns=20000, H=40, K=15, Cin=128, Cout=128
// out[n,d] = sum_k sum_c ( sum_h clip(1-|nb(n,h)-kp_k|/1.2,0) * x[nb(n,h),c] ) * W[k,c,d]

typedef __attribute__((ext_vector_type(2))) float v2f;
typedef __attribute__((ext_vector_type(8))) float v8f;

constexpr int   Hn     = 40;    // neighbors
constexpr int   Kp     = 15;    // kernel points (padded to 16 in LDS)
constexpr int   CIN    = 128;
constexpr int   COUT   = 128;
constexpr int   PTS    = 16;    // query points per workgroup (M-tile of stage C)
constexpr int   HPAD   = 44;    // h-stride in s_w: 44*l mod 64 distinct -> conflict-free A reads
constexpr int   PSTRIDE = 132;  // p-stride in s_wt: 132 mod 64 = 4 -> conflict-free A reads
constexpr int   KSTRIDE = PTS * PSTRIDE + 4; // k-plane stride; +4 de-conflicts half-wave stores
constexpr float KP_EXT_INV = 1.0f / 1.2f;

__global__ __launch_bounds__(256)
void kpconv_wmma_kernel(const float* __restrict__ q_pts,
                        const float* __restrict__ s_pts,
                        const int*   __restrict__ nbi,
                        const float* __restrict__ x,
                        const float* __restrict__ kpts,
                        const float* __restrict__ wts,
                        float*       __restrict__ out,
                        int N, int Ns)
{
    __shared__ int   s_idx[PTS * Hn];          //  2.5 KB  neighbor indices
    __shared__ float s_rel[PTS * Hn * 3];      //  7.5 KB  centered neighbor coords
    __shared__ float s_kp [16 * 3];            //  tiny    kernel point coords
    __shared__ float s_w  [PTS * 16 * HPAD];   // 44.0 KB  influence weights [p][k][h]
    __shared__ float s_wt [Kp * KSTRIDE];      //124.0 KB  weighted features [k][p][c]

    const int tid = threadIdx.x;
    const int n0  = blockIdx.x * PTS;

    if (tid < 16 * 3) s_kp[tid] = (tid < Kp * 3) ? kpts[tid] : 0.0f;

    // ---- Stage A1: gather neighbor indices + centered coords -------------
    for (int i = tid; i < PTS * Hn; i += 256) {
        int p = i / Hn, h = i - p * Hn;
        int n = n0 + p;
        int idx = 0;
        float rx = 0.f, ry = 0.f, rz = 0.f;
        if (n < N) {
            idx = nbi[n * Hn + h];
            int  ic   = (idx < Ns) ? idx : (Ns - 1);
            bool real = (idx < Ns);                  // shadow guard (ref pads with 1e6)
            float sx = real ? s_pts[ic * 3 + 0] : 1.0e6f;
            float sy = real ? s_pts[ic * 3 + 1] : 1.0e6f;
            float sz = real ? s_pts[ic * 3 + 2] : 1.0e6f;
            rx = sx - q_pts[n * 3 + 0];
            ry = sy - q_pts[n * 3 + 1];
            rz = sz - q_pts[n * 3 + 2];
            idx = ic;
        }
        s_idx[i] = idx;
        s_rel[i * 3 + 0] = rx;
        s_rel[i * 3 + 1] = ry;
        s_rel[i * 3 + 2] = rz;
    }
    __syncthreads();

    // ---- Stage A2: influence weights w[p][k][h] (zero-padded) ------------
    for (int i = tid; i < PTS * 16 * HPAD; i += 256) {
        int p = i / (16 * HPAD);
        int r = i - p * 16 * HPAD;
        int k = r / HPAD;
        int h = r - k * HPAD;
        float w = 0.0f;
        if (k < Kp && h < Hn && (n0 + p) < N) {
            int   b  = (p * Hn + h) * 3;
            float dx = s_rel[b + 0] - s_kp[k * 3 + 0];
            float dy = s_rel[b + 1] - s_kp[k * 3 + 1];
            float dz = s_rel[b + 2] - s_kp[k * 3 + 2];
            float d  = sqrtf(dx * dx + dy * dy + dz * dz);
            w = fmaxf(0.0f, 1.0f - d * KP_EXT_INV);
        }
        s_w[i] = w;   // i == (p*16 + k)*HPAD + h
    }
    __syncthreads();

    const int wv   = tid >> 5;         // wave id 0..7 (wave32)
    const int lane = tid & 31;
    const int half = lane >> 4;        // 0: lanes 0-15, 1: lanes 16-31
    const int l15  = lane & 15;
    const int hoff = half << 1;        // K-dim offset of A/B fragments (0 or 2)

    // ---- Stage B: weighted[p][k][c] = sum_h w[p][k][h] * x[nb(p,h)][c] ---
    // per wave: 2 points, each [16k x 40h] x [40h x 128c] via wmma f32 16x16x4
    for (int pp = 0; pp < 2; ++pp) {
        int p = wv * 2 + pp;
        const float* wrow = &s_w[(p * 16 + l15) * HPAD];  // A row M=l15 (kernel pt)
        const int*   irow = &s_idx[p * Hn];
        for (int nt = 0; nt < 8; ++nt) {
            int c = nt * 16 + l15;                        // B/D column
            v8f acc = {};
#pragma unroll
            for (int hs = 0; hs < Hn / 4; ++hs) {
                int h0 = hs * 4 + hoff;
                v2f a, b;
                a.x = wrow[h0];
                a.y = wrow[h0 + 1];
                b.x = x[irow[h0]     * CIN + c];
                b.y = x[irow[h0 + 1] * CIN + c];
                acc = __builtin_amdgcn_wmma_f32_16x16x4_f32(
                          false, a, false, b, (short)0, acc, false, false);
            }
#pragma unroll
            for (int j = 0; j < 8; ++j) {
                int k = j + half * 8;                     // D row M = j + 8*half
                if (k < Kp)
                    s_wt[k * KSTRIDE + p * PSTRIDE + nt * 16 + l15] = acc[j];
            }
        }
    }
    __syncthreads();

    // ---- Stage C: out[p][d] = sum_k wt[k][p][:] x W[k][:][d] -------------
    // per wave: one 16-wide Cout tile, M=16 points, K = 15*128
    const int d0 = wv * 16;
    v8f acc = {};
    for (int k = 0; k < Kp; ++k) {
        const float* Wk   = wts + (size_t)k * CIN * COUT;
        const float* arow = &s_wt[k * KSTRIDE + l15 * PSTRIDE]; // A row M=l15 (point)
#pragma unroll 8
        for (int cs = 0; cs < CIN / 4; ++cs) {
            int cc = cs * 4 + hoff;
            v2f a, b;
            a.x = arow[cc];
            a.y = arow[cc + 1];
            b.x = Wk[(cc)     * COUT + d0 + l15];
            b.y = Wk[(cc + 1) * COUT + d0 + l15];
            acc = __builtin_amdgcn_wmma_f32_16x16x4_f32(
                      false, a, false, b, (short)0, acc, false, false);
        }
    }
#pragma unroll
    for (int j = 0; j < 8; ++j) {
        int p = j + half * 8;
        int n = n0 + p;
        if (n < N) out[(size_t)n * COUT + d0 + l15] = acc[j];
    }
}

extern "C" void kernel_launch(void* const* d_in, const int* in_sizes, int n_in,
                              void* d_out, int out_size, void* d_ws, size_t ws_size,
                              hipStream_t stream) {
    const float* q_pts = (const float*)d_in[0];
    const float* s_pts = (const float*)d_in[1];
    const int*   nbi   = (const int*)  d_in[2];
    const float* x     = (const float*)d_in[3];
    const float* kpts  = (const float*)d_in[4];
    const float* wts   = (const float*)d_in[5];
    float*       out   = (float*)d_out;

    int N  = in_sizes[0] / 3;
    int Ns = in_sizes[1] / 3;

    int grid = (N + PTS - 1) / PTS;
    hipLaunchKernelGGL(kpconv_wmma_kernel, dim3(grid), dim3(256), 0, stream,
                       q_pts, s_pts, nbi, x, kpts, wts, out, N, Ns);
}